// ImageSelfAttention_74526272520432
// MI455X (gfx1250) — compile-verified
//
#include <hip/hip_runtime.h>
#include <hip/hip_bf16.h>

typedef __attribute__((ext_vector_type(2))) float v2f;
typedef __attribute__((ext_vector_type(4))) float v4f;
typedef __attribute__((ext_vector_type(8))) float v8f;

#define BATCH 4
#define CH    256
#define NN    4096
#define INTER 32

#if defined(__has_builtin)
#if __has_builtin(__builtin_amdgcn_permlane16)
#define USE_PERMLANE16 1
#endif
#endif

// permute within each 16-lane row of the wave32: lane i reads lane sel[i]
__device__ __forceinline__ float pl16(float v, unsigned s0, unsigned s1)
{
#ifdef USE_PERMLANE16
    unsigned x = __float_as_uint(v);
    x = __builtin_amdgcn_permlane16(x, x, s0, s1, false, false);
    return __uint_as_float(x);
#else
    return v;
#endif
}

// max across each 16-lane half (xor butterfly, pure VALU via v_permlane16_b32)
__device__ __forceinline__ float red_max16(float v)
{
#ifdef USE_PERMLANE16
    v = fmaxf(v, pl16(v, 0x67452301u, 0xEFCDAB89u)); // ^1
    v = fmaxf(v, pl16(v, 0x54761032u, 0xDCFE98BAu)); // ^2
    v = fmaxf(v, pl16(v, 0x32107654u, 0xBA98FEDCu)); // ^4
    v = fmaxf(v, pl16(v, 0xFEDCBA98u, 0x76543210u)); // ^8
#else
    v = fmaxf(v, __shfl_xor(v, 1, 32));
    v = fmaxf(v, __shfl_xor(v, 2, 32));
    v = fmaxf(v, __shfl_xor(v, 4, 32));
    v = fmaxf(v, __shfl_xor(v, 8, 32));
#endif
    return v;
}

__device__ __forceinline__ float red_sum16(float v)
{
#ifdef USE_PERMLANE16
    v += pl16(v, 0x67452301u, 0xEFCDAB89u);
    v += pl16(v, 0x54761032u, 0xDCFE98BAu);
    v += pl16(v, 0x32107654u, 0xBA98FEDCu);
    v += pl16(v, 0xFEDCBA98u, 0x76543210u);
#else
    v += __shfl_xor(v, 1, 32);
    v += __shfl_xor(v, 2, 32);
    v += __shfl_xor(v, 4, 32);
    v += __shfl_xor(v, 8, 32);
#endif
    return v;
}

// ---------------------------------------------------------------------------
// Kernel 1: 1x1-conv projections.  Weights staged transposed in LDS so each
// c-step is 8 broadcast ds_load_b128 feeding 32 FMAs.  x read coalesced in n.
// ---------------------------------------------------------------------------
__global__ __launch_bounds__(256) void proj_kernel(
    const float* __restrict__ x,
    const float* __restrict__ Wq, const float* __restrict__ bq,
    const float* __restrict__ Wk, const float* __restrict__ bk,
    const float* __restrict__ Wv, const float* __restrict__ bv,
    float* __restrict__ Qt, float* __restrict__ Kt, float* __restrict__ Vt)
{
    __shared__ float Wst[256 * 32];     // [c][o] transposed
    __shared__ float bs[32];

    const int g = blockIdx.y;   // 0:q  1:k  2..9: v chunk
    const int b = blockIdx.z;

    const float* W;
    const float* bias;
    int o0;
    float scale;
    if (g == 0)      { W = Wq;                      bias = bq;                o0 = 0;            scale = 0.17677669529663687f; }
    else if (g == 1) { W = Wk;                      bias = bk;                o0 = 0;            scale = 1.0f; }
    else             { W = Wv + (g - 2) * 32 * 256; bias = bv + (g - 2) * 32; o0 = (g - 2) * 32; scale = 1.0f; }

    for (int i = threadIdx.x; i < 32 * 256; i += 256) {
        const int o = i >> 8, c = i & 255;          // W is [o][c]
        Wst[c * 32 + o] = W[i];
    }
    if (threadIdx.x < 32) bs[threadIdx.x] = bias[threadIdx.x];
    __syncthreads();

    const int n = blockIdx.x * 256 + threadIdx.x;
    v4f acc[8];
#pragma unroll
    for (int q = 0; q < 8; ++q) acc[q] = (v4f){0.f, 0.f, 0.f, 0.f};

    const float* xb = x + (size_t)b * CH * NN + n;
    for (int c = 0; c < CH; ++c) {
        const float xv = xb[(size_t)c * NN];
        const v4f* wr4 = (const v4f*)&Wst[c * 32];
#pragma unroll
        for (int q = 0; q < 8; ++q) acc[q] += wr4[q] * xv;
    }

    if (g < 2) {
        float* dst = (g == 0 ? Qt : Kt) + ((size_t)(b * NN + n)) * INTER;
#pragma unroll
        for (int q = 0; q < 8; ++q)
#pragma unroll
            for (int e = 0; e < 4; ++e) dst[q * 4 + e] = (acc[q][e] + bs[q * 4 + e]) * scale;
    } else {
        float* dst = Vt + ((size_t)(b * NN + n)) * CH + o0;
#pragma unroll
        for (int q = 0; q < 8; ++q)
#pragma unroll
            for (int e = 0; e < 4; ++e) dst[q * 4 + e] = acc[q][e] + bs[q * 4 + e];
    }
}

// ---------------------------------------------------------------------------
// Kernel 2: flash attention, f32 WMMA (16x16x4).
// Block = 64 query rows, 512 threads = 16 waves as 4 row-groups x 4 col-groups.
// ---------------------------------------------------------------------------
__global__ __launch_bounds__(512) void attn_kernel(
    const float* __restrict__ x, const float* __restrict__ gamma,
    const float* __restrict__ Qt, const float* __restrict__ Kt,
    const float* __restrict__ Vt, float* __restrict__ out)
{
    __shared__ float Qs[64][36];
    __shared__ float Ks[64][36];
    __shared__ float Vs[64][260];
    __shared__ float Ps[64][68];
    __shared__ float m_s[64], l_s[64], alpha_s[64];
    __shared__ float pmax[64][4], psum[64][4];

    const int tid  = threadIdx.x;
    const int lane = tid & 31;
    const int wave = tid >> 5;
    const int wr   = wave >> 2;          // query row group (16 rows)
    const int wc   = wave & 3;           // col group
    const int hl   = (lane < 16) ? 0 : 1;
    const int l15  = lane & 15;
    const int b     = blockIdx.y;
    const int nbase = blockIdx.x * 64;
    const int qr0   = wr * 16;

    for (int i = tid; i < 64 * 32; i += 512)
        Qs[i >> 5][i & 31] = Qt[((size_t)(b * NN + nbase)) * INTER + i];
    if (tid < 64) { m_s[tid] = -1e30f; l_s[tid] = 0.0f; }

    v8f acc0 = {}, acc1 = {}, acc2 = {}, acc3 = {};

    for (int kt = 0; kt < NN / 64; ++kt) {
        const int ko = kt * 64;
        __syncthreads();               // protect Ks/Vs/Ps reuse
        for (int i = tid; i < 64 * 32; i += 512)
            Ks[i >> 5][i & 31] = Kt[((size_t)(b * NN + ko)) * INTER + i];
        for (int i = tid; i < 64 * 256; i += 512)
            Vs[i >> 8][i & 255] = Vt[((size_t)(b * NN + ko)) * CH + i];

        // prefetch next tile into cache while this one computes
        if (kt + 1 < NN / 64) {
            const float* nV = Vt + ((size_t)(b * NN + ko + 64)) * CH;
            __builtin_prefetch(nV + (size_t)tid * 32, 0, 1);     // 512 x 128B = 64KB
            if (tid < 64) {
                const float* nK = Kt + ((size_t)(b * NN + ko + 64)) * INTER;
                __builtin_prefetch(nK + (size_t)tid * 32, 0, 1); // 64 x 128B = 8KB
            }
        }
        __syncthreads();

        // ---- S = Q K^T : one 16x16 tile per wave, 8 k-steps of 4 ----
        v8f s = {};
#pragma unroll
        for (int kk = 0; kk < 8; ++kk) {
            const int k0 = kk * 4 + hl * 2;
            v2f a;  a.x  = Qs[qr0 + l15][k0];      a.y  = Qs[qr0 + l15][k0 + 1];
            v2f bb; bb.x = Ks[wc * 16 + l15][k0];  bb.y = Ks[wc * 16 + l15][k0 + 1];
            s = __builtin_amdgcn_wmma_f32_16x16x4_f32(false, a, false, bb,
                                                      (short)0, s, false, false);
        }

        // ---- per-row max of this 16-col chunk (VALU permlane reductions) ----
        float rmax[8];
#pragma unroll
        for (int j = 0; j < 8; ++j) rmax[j] = red_max16(s[j]);
        if (l15 == 0) {
#pragma unroll
            for (int j = 0; j < 8; ++j) pmax[qr0 + j + hl * 8][wc] = rmax[j];
        }
        __syncthreads();

        if (wc == 0 && lane < 16) {    // combine 4 partials, update running max
            const int row = qr0 + lane;
            float tm = fmaxf(fmaxf(pmax[row][0], pmax[row][1]),
                             fmaxf(pmax[row][2], pmax[row][3]));
            float mo = m_s[row];
            float mn = fmaxf(mo, tm);
            m_s[row]     = mn;
            alpha_s[row] = __expf(mo - mn);
        }
        __syncthreads();

        // ---- P = exp(S - m), stage to LDS for PV, partial row sums ----
        float rsum[8];
#pragma unroll
        for (int j = 0; j < 8; ++j) {
            const int row = qr0 + j + hl * 8;
            float p = __expf(s[j] - m_s[row]);
            Ps[row][wc * 16 + l15] = p;
            rsum[j] = red_sum16(p);
        }
        if (l15 == 0) {
#pragma unroll
            for (int j = 0; j < 8; ++j) psum[qr0 + j + hl * 8][wc] = rsum[j];
        }
        __syncthreads();

        if (wc == 0 && lane < 16) {
            const int row = qr0 + lane;
            l_s[row] = l_s[row] * alpha_s[row] +
                       (psum[row][0] + psum[row][1] + psum[row][2] + psum[row][3]);
        }

        // ---- rescale accumulators, then PV: (16x64) x (64 x 64-cols) ----
        float al[8];
#pragma unroll
        for (int j = 0; j < 8; ++j) al[j] = alpha_s[qr0 + j + hl * 8];
#pragma unroll
        for (int j = 0; j < 8; ++j) {
            acc0[j] *= al[j]; acc1[j] *= al[j]; acc2[j] *= al[j]; acc3[j] *= al[j];
        }

        const int cb = wc * 64 + l15;
#pragma unroll 4
        for (int kk = 0; kk < 16; ++kk) {
            const int k0 = kk * 4 + hl * 2;
            v2f a;  a.x = Ps[qr0 + l15][k0]; a.y = Ps[qr0 + l15][k0 + 1];
            v2f b0; b0.x = Vs[k0][cb];      b0.y = Vs[k0 + 1][cb];
            v2f b1; b1.x = Vs[k0][cb + 16]; b1.y = Vs[k0 + 1][cb + 16];
            v2f b2; b2.x = Vs[k0][cb + 32]; b2.y = Vs[k0 + 1][cb + 32];
            v2f b3; b3.x = Vs[k0][cb + 48]; b3.y = Vs[k0 + 1][cb + 48];
            acc0 = __builtin_amdgcn_wmma_f32_16x16x4_f32(false, a, false, b0, (short)0, acc0, false, false);
            acc1 = __builtin_amdgcn_wmma_f32_16x16x4_f32(false, a, false, b1, (short)0, acc1, false, false);
            acc2 = __builtin_amdgcn_wmma_f32_16x16x4_f32(false, a, false, b2, (short)0, acc2, false, false);
            acc3 = __builtin_amdgcn_wmma_f32_16x16x4_f32(false, a, false, b3, (short)0, acc3, false, false);
        }
    }

    // ---- epilogue: normalize, transpose via LDS, y = gamma*out + x ----
    __syncthreads();
    float rl[8];
#pragma unroll
    for (int j = 0; j < 8; ++j) rl[j] = 1.0f / l_s[qr0 + j + hl * 8];
#pragma unroll
    for (int j = 0; j < 8; ++j) {
        const int nn = qr0 + j + hl * 8;
        const int cc = wc * 64 + l15;
        Vs[nn][cc]      = acc0[j] * rl[j];
        Vs[nn][cc + 16] = acc1[j] * rl[j];
        Vs[nn][cc + 32] = acc2[j] * rl[j];
        Vs[nn][cc + 48] = acc3[j] * rl[j];
    }
    __syncthreads();

    const float gm = gamma[0];
    for (int i = tid; i < 64 * 256; i += 512) {
        const int c  = i >> 6;
        const int nn = i & 63;
        const size_t idx = ((size_t)(b * CH + c)) * NN + nbase + nn;
        out[idx] = gm * Vs[nn][c] + x[idx];
    }
}

// ---------------------------------------------------------------------------
extern "C" void kernel_launch(void* const* d_in, const int* in_sizes, int n_in,
                              void* d_out, int out_size, void* d_ws, size_t ws_size,
                              hipStream_t stream)
{
    const float* x     = (const float*)d_in[0];
    const float* Wq    = (const float*)d_in[1];
    const float* bq    = (const float*)d_in[2];
    const float* Wk    = (const float*)d_in[3];
    const float* bk    = (const float*)d_in[4];
    const float* Wv    = (const float*)d_in[5];
    const float* bv    = (const float*)d_in[6];
    const float* gamma = (const float*)d_in[7];
    float* out = (float*)d_out;

    float* ws = (float*)d_ws;
    float* Qt = ws;                                   // [B,N,32]
    float* Kt = Qt + (size_t)BATCH * NN * INTER;      // [B,N,32]
    float* Vt = Kt + (size_t)BATCH * NN * INTER;      // [B,N,256]

    dim3 g1(NN / 256, 10, BATCH);
    proj_kernel<<<g1, 256, 0, stream>>>(x, Wq, bq, Wk, bk, Wv, bv, Qt, Kt, Vt);

    dim3 g2(NN / 64, BATCH);
    attn_kernel<<<g2, 512, 0, stream>>>(x, gamma, Qt, Kt, Vt, out);
}